// ProteinInteractionGNN_24764781428928
// MI455X (gfx1250) — compile-verified
//
#include <hip/hip_runtime.h>
#include <hip/hip_bf16.h>
#include <math.h>

typedef float v2f __attribute__((ext_vector_type(2)));
typedef float v8f __attribute__((ext_vector_type(8)));

#define NEG_SLOPE 0.01f

// ---------------------------------------------------------------------------
// generic u32 fill (used for zeroing f32 buffers and seeding the max-pool
// buffer with the encoding of -inf)
// ---------------------------------------------------------------------------
__global__ void fill_u32_kernel(unsigned* __restrict__ p, unsigned val, size_t n) {
    size_t i = (size_t)blockIdx.x * blockDim.x + threadIdx.x;
    size_t stride = (size_t)gridDim.x * blockDim.x;
    for (; i < n; i += stride) p[i] = val;
}

// ---------------------------------------------------------------------------
// in-degree of destination nodes (float accumulate; self-loop added later)
// ---------------------------------------------------------------------------
__global__ void degree_kernel(const int* __restrict__ cols, float* __restrict__ deg, int ne) {
    int i = blockIdx.x * blockDim.x + threadIdx.x;
    int stride = gridDim.x * blockDim.x;
    for (; i < ne; i += stride) atomicAdd(&deg[cols[i]], 1.0f);
}

// deg -> dinv = rsqrt(deg + 1)   (+1 = self loop, guarantees deg > 0)
__global__ void dinv_kernel(float* __restrict__ deg, int n) {
    int i = blockIdx.x * blockDim.x + threadIdx.x;
    if (i < n) deg[i] = rsqrtf(deg[i] + 1.0f);
}

// ---------------------------------------------------------------------------
// XW = X[N,128] @ W[128,128] using full-precision f32 WMMA (16x16x4).
// 256 threads = 8 waves; each wave computes a 16x128 output strip
// (8 accumulator tiles). W is staged in 64KB LDS once per block.
//
// A-frag (16x4 f32, ISA 7.12.2): lane m = lane&15 holds row M=m;
//   half h = lane>>4: VGPR0 = K = 2h, VGPR1 = K = 2h+1.
// C/D (16x16 f32): VGPR r, lanes 0-15 -> M=r, lanes 16-31 -> M=r+8, N=lane&15.
// ---------------------------------------------------------------------------
__global__ __launch_bounds__(256) void xw_wmma_kernel(const float* __restrict__ X,
                                                      const float* __restrict__ W,
                                                      float* __restrict__ XW,
                                                      int n_rows) {
    __shared__ float Wlds[128 * 128];  // 64 KB of the 320 KB WGP LDS
    int tid = threadIdx.x;
    for (int i = tid; i < 128 * 128; i += 256) Wlds[i] = W[i];
    __syncthreads();

    int wave = tid >> 5;
    int lane = tid & 31;
    int row0 = (blockIdx.x * 8 + wave) * 16;
    if (row0 >= n_rows) return;              // wave-uniform: EXEC stays all-ones

    int m = lane & 15;
    int h = lane >> 4;
    int row = row0 + m;
    if (row >= n_rows) row = n_rows - 1;     // branchless clamp for tail tile

    v8f acc[8] = {};
    const float* xrow = X + (size_t)row * 128;

    for (int k = 0; k < 128; k += 4) {
        v2f a;
        a.x = xrow[k + 2 * h];
        a.y = xrow[k + 2 * h + 1];
#pragma unroll
        for (int j = 0; j < 8; ++j) {
            int n = j * 16 + m;
            v2f b;
            b.x = Wlds[(k + 2 * h) * 128 + n];
            b.y = Wlds[(k + 2 * h + 1) * 128 + n];
            // (neg_a, A, neg_b, B, c_mod, C, reuse_a, reuse_b)
            acc[j] = __builtin_amdgcn_wmma_f32_16x16x4_f32(
                false, a, false, b, (short)0, acc[j], false, false);
        }
    }

#pragma unroll
    for (int r = 0; r < 8; ++r) {
        int outrow = row0 + r + 8 * h;
        if (outrow < n_rows) {
            float* orow = XW + (size_t)outrow * 128;
#pragma unroll
            for (int j = 0; j < 8; ++j) orow[j * 16 + m] = acc[j][r];
        }
    }
}

// ---------------------------------------------------------------------------
// edge scatter: one wave per edge; lane loads a float4 of xw[row] and
// atomically adds norm * v into agg[col]. xw+agg (102 MB) sit in L2 (192 MB).
// ---------------------------------------------------------------------------
__global__ void edge_scatter_kernel(const int* __restrict__ rows,
                                    const int* __restrict__ cols,
                                    const float* __restrict__ dinv,
                                    const float* __restrict__ xw,
                                    float* __restrict__ agg, int ne) {
    int gwave = (blockIdx.x * blockDim.x + threadIdx.x) >> 5;
    int lane = threadIdx.x & 31;
    int nwaves = (gridDim.x * blockDim.x) >> 5;
    for (int e = gwave; e < ne; e += nwaves) {
        int r = rows[e];
        int c = cols[e];
        float nm = dinv[r] * dinv[c];
        const float4* src = (const float4*)(xw + (size_t)r * 128);
        float* dst = agg + (size_t)c * 128 + lane * 4;
        float4 v = src[lane];
        atomicAdd(dst + 0, nm * v.x);
        atomicAdd(dst + 1, nm * v.y);
        atomicAdd(dst + 2, nm * v.z);
        atomicAdd(dst + 3, nm * v.w);
    }
}

// order-preserving float<->uint encoding for atomicMax-based segment_max
__device__ __forceinline__ unsigned enc_f32(float f) {
    unsigned u = __float_as_uint(f);
    return (u & 0x80000000u) ? ~u : (u | 0x80000000u);
}
__device__ __forceinline__ float dec_f32(unsigned e) {
    unsigned u = (e & 0x80000000u) ? (e ^ 0x80000000u) : ~e;
    return __uint_as_float(u);
}

// ---------------------------------------------------------------------------
// fused epilogue: h = lrelu(agg + dinv^2 * xw + bias); pooled directly into
// genc[batch, :] via encoded atomicMax (h never written to memory).
// thread = (node, 4 features)
// ---------------------------------------------------------------------------
__global__ void epilogue_pool_kernel(const float* __restrict__ agg,
                                     const float* __restrict__ xw,
                                     const float* __restrict__ dinv,
                                     const float* __restrict__ bias,
                                     const int* __restrict__ batch,
                                     unsigned* __restrict__ genc, int n) {
    size_t t = (size_t)blockIdx.x * blockDim.x + threadIdx.x;
    size_t total = (size_t)n * 32;
    size_t stride = (size_t)gridDim.x * blockDim.x;
    for (; t < total; t += stride) {
        int node = (int)(t >> 5);
        int f4 = ((int)t & 31) * 4;
        float di = dinv[node];
        float sl = di * di;
        const float* a = agg + (size_t)node * 128 + f4;
        const float* x = xw + (size_t)node * 128 + f4;
        unsigned* g = genc + (size_t)batch[node] * 128 + f4;
#pragma unroll
        for (int u = 0; u < 4; ++u) {
            float hv = a[u] + sl * x[u] + bias[f4 + u];
            hv = (hv >= 0.0f) ? hv : NEG_SLOPE * hv;
            atomicMax(g + u, enc_f32(hv));
        }
    }
}

__global__ void decode_kernel(const unsigned* __restrict__ e, float* __restrict__ out, int n) {
    int i = blockIdx.x * blockDim.x + threadIdx.x;
    if (i < n) out[i] = dec_f32(e[i]);
}

// ---------------------------------------------------------------------------
// small dense FC: out[i,j] = act(sum_k A[i,k] * W[k,j] + b[j])
// act: 0=id, 1=leaky_relu, 2=sigmoid.  Used only for [128 x *] tails.
// ---------------------------------------------------------------------------
__global__ void fc_kernel(const float* __restrict__ A, int lda,
                          const float* __restrict__ W, const float* __restrict__ bias,
                          float* __restrict__ out, int ldo,
                          int M, int K, int Nc, int act) {
    int idx = blockIdx.x * blockDim.x + threadIdx.x;
    if (idx >= M * Nc) return;
    int i = idx / Nc;
    int j = idx - i * Nc;
    float s = bias[j];
    for (int k = 0; k < K; ++k) s += A[(size_t)i * lda + k] * W[(size_t)k * Nc + j];
    if (act == 1) s = (s >= 0.0f) ? s : NEG_SLOPE * s;
    else if (act == 2) s = 1.0f / (1.0f + expf(-s));
    out[(size_t)i * ldo + j] = s;
}

// ---------------------------------------------------------------------------
extern "C" void kernel_launch(void* const* d_in, const int* in_sizes, int n_in,
                              void* d_out, int out_size, void* d_ws, size_t ws_size,
                              hipStream_t stream) {
    const float* pro_x[2]  = {(const float*)d_in[0], (const float*)d_in[1]};
    const int*   pro_ei[2] = {(const int*)d_in[2], (const int*)d_in[3]};
    const int*   pro_b[2]  = {(const int*)d_in[4], (const int*)d_in[5]};
    const float* convW[2]  = {(const float*)d_in[6], (const float*)d_in[8]};
    const float* convb[2]  = {(const float*)d_in[7], (const float*)d_in[9]};
    const float* fcpW[2]   = {(const float*)d_in[10], (const float*)d_in[12]};
    const float* fcpb[2]   = {(const float*)d_in[11], (const float*)d_in[13]};
    const float* fc1W = (const float*)d_in[14];
    const float* fc1b = (const float*)d_in[15];
    const float* fc2W = (const float*)d_in[16];
    const float* fc2b = (const float*)d_in[17];
    const float* outW = (const float*)d_in[18];
    const float* outb = (const float*)d_in[19];

    const int D = 128;
    int N = in_sizes[0] / D;       // 100000
    int E = in_sizes[2] / 2;       // 1600000
    int B = out_size;              // 128 graphs

    // workspace layout (branches run sequentially -> reuse big buffers;
    // keeps xw+agg = 102 MB resident in the 192 MB L2 during edge scatter)
    char* ws = (char*)d_ws;
    float*    xw   = (float*)ws;    ws += (size_t)N * D * 4;
    float*    agg  = (float*)ws;    ws += (size_t)N * D * 4;
    float*    dinv = (float*)ws;    ws += (size_t)N * 4;
    unsigned* genc = (unsigned*)ws; ws += (size_t)B * D * 4;
    float*    gdec = (float*)ws;    ws += (size_t)B * D * 4;
    float*    cbuf = (float*)ws;    ws += (size_t)B * 2 * D * 4;  // [B, 256] concat
    float*    c1   = (float*)ws;    ws += (size_t)B * 256 * 4;
    float*    c2   = (float*)ws;    ws += (size_t)B * 64 * 4;

    const unsigned ENC_NEG_INF = 0x007FFFFFu;  // enc_f32(-inf)

    for (int br = 0; br < 2; ++br) {
        const int* erow = pro_ei[br];        // edge_index[0] = sources
        const int* ecol = pro_ei[br] + E;    // edge_index[1] = destinations

        fill_u32_kernel<<<2048, 256, 0, stream>>>((unsigned*)dinv, 0u, (size_t)N);
        fill_u32_kernel<<<4096, 256, 0, stream>>>((unsigned*)agg, 0u, (size_t)N * D);
        fill_u32_kernel<<<64, 256, 0, stream>>>(genc, ENC_NEG_INF, (size_t)B * D);

        degree_kernel<<<2048, 256, 0, stream>>>(ecol, dinv, E);
        dinv_kernel<<<(N + 255) / 256, 256, 0, stream>>>(dinv, N);

        xw_wmma_kernel<<<(N + 127) / 128, 256, 0, stream>>>(pro_x[br], convW[br], xw, N);

        edge_scatter_kernel<<<4096, 256, 0, stream>>>(erow, ecol, dinv, xw, agg, E);

        epilogue_pool_kernel<<<4096, 256, 0, stream>>>(agg, xw, dinv, convb[br],
                                                       pro_b[br], genc, N);

        decode_kernel<<<(B * D + 255) / 256, 256, 0, stream>>>(genc, gdec, B * D);

        // g @ fc_p_W + b, lrelu -> columns [br*128, br*128+128) of cbuf
        fc_kernel<<<(B * D + 255) / 256, 256, 0, stream>>>(
            gdec, D, fcpW[br], fcpb[br], cbuf + br * D, 2 * D, B, D, D, 1);
    }

    // head MLP
    fc_kernel<<<(B * 256 + 255) / 256, 256, 0, stream>>>(
        cbuf, 256, fc1W, fc1b, c1, 256, B, 256, 256, 1);
    fc_kernel<<<(B * 64 + 255) / 256, 256, 0, stream>>>(
        c1, 256, fc2W, fc2b, c2, 64, B, 256, 64, 1);
    fc_kernel<<<(B + 255) / 256, 256, 0, stream>>>(
        c2, 64, outW, outb, (float*)d_out, 1, B, 64, 1, 2);
}